// PHY_70935679861195
// MI455X (gfx1250) — compile-verified
//
#include <hip/hip_runtime.h>
#include <hip/hip_bf16.h>

#define NTIPS   128
#define SEQLEN  2000
#define BS      32
#define NINT    (NTIPS - 2)       // 126
#define NNODES  (2 * NTIPS - 2)   // 254
#define TILES   (SEQLEN / 16)     // 125 site-tiles of 16
#define NPAIRS  ((TILES + 1) / 2) // 63 tile-pairs (32 sites per wave)
#define TSLOTS  128               // padded tile slots in partials
#define LN2F    0.6931471805599453f

typedef __attribute__((ext_vector_type(2))) float v2f;
typedef __attribute__((ext_vector_type(8))) float v8f;

#define RFL(x) __builtin_amdgcn_readfirstlane((int)(x))

// ---------------------------------------------------------------------------
// Kernel 1: precompute T[b][n] = max(U @ diag(exp(exp(lb)*D)) @ Uinv, 0)
// 32*254 = 8128 tiny 4x4 matrices -> 520 KB in d_ws (L2 resident).
// ---------------------------------------------------------------------------
__global__ void phy_Tprep(const float* __restrict__ log_branch,
                          const float* __restrict__ D4,
                          const float* __restrict__ U,
                          const float* __restrict__ Uinv,
                          float* __restrict__ T) {
    int id = blockIdx.x * blockDim.x + threadIdx.x;
    if (id >= BS * NNODES) return;
    float t = expf(log_branch[id]);
    float e[4];
#pragma unroll
    for (int k = 0; k < 4; ++k) e[k] = expf(t * D4[k]);
#pragma unroll
    for (int i = 0; i < 4; ++i) {
#pragma unroll
        for (int j = 0; j < 4; ++j) {
            float s = 0.f;
#pragma unroll
            for (int k = 0; k < 4; ++k) s += U[i * 4 + k] * e[k] * Uinv[k * 4 + j];
            T[id * 16 + i * 4 + j] = fmaxf(s, 0.f);
        }
    }
}

// ---------------------------------------------------------------------------
// Kernel 2: sequential pruning scan. One wave owns (batch b, 2 tiles of 16
// sites). Per child: D = T(A, rows clamped m&3) @ ll(B) via
// V_WMMA_F32_16X16X4_F32. Row clamping makes lane halves duplicates, so the
// scaler and the next-step rolling B operand are lane-local.
// Topology scalars pipelined 2 deep; T / tip-L vector loads pipelined 1 deep.
// Recurrence: WMMA -> (pk_mul) -> pk_add -> v_rcp -> cndmask+mul -> WMMA.
// WMMA 32-bit operand layouts (ISA 7.12.2), lane l, h=l>>4, n=l&15:
//   A 16x4 : v0=A[n][2h], v1=A[n][2h+1]          (A[n] := T[n&3])
//   B 4x16 : v0=B[2h][n], v1=B[2h+1][n]          (B[k][n] = ll[k][site n])
//   D      : vgpr r holds (T@ll)[r][n] for r=0..3, duplicated across halves
// ---------------------------------------------------------------------------
__global__ __launch_bounds__(256) void phy_scan(
    const float* __restrict__ Tbuf,               // [BS][NNODES][4][4]
    const float* __restrict__ L,                  // [NTIPS][4][SEQLEN]
    const float* __restrict__ site_counts,        // [SEQLEN]
    const float* __restrict__ pden,               // [4]
    const int*  __restrict__ children,            // [NINT][BS][3]
    const unsigned char* __restrict__ child_mask, // [NINT][3]
    const int*  __restrict__ parent,              // [NINT][BS]
    float* __restrict__ partials) {               // [BS][TSLOTS]
    const int b    = blockIdx.y;
    const int wave = threadIdx.x >> 5;
    const int lane = threadIdx.x & 31;
    const int pair = blockIdx.x * 8 + wave;
    if (pair >= NPAIRS) return;                   // whole wave exits together
    const int  tA   = pair * 2;
    const int  tB   = tA + 1;
    const bool hasB = (tB < TILES);
    const int  sA   = tA * 16;
    const int  sB   = hasB ? tB * 16 : (SEQLEN - 16); // clamp: dup work, dropped

    const int n  = lane & 15;                     // site within tile / D column
    const int h  = lane >> 4;
    const int kb = h << 1;                        // K pair owned by this half
    const int ar = lane & 3;                      // clamped T row for A operand

    // loop-invariant per-lane element offsets (scalar base added per child)
    const int offT  = ar * 4 + kb;                // into a 4x4 T block
    const int offLA = kb * SEQLEN + sA + n;       // into a tip's 4xSEQLEN block
    const int offLB = kb * SEQLEN + sB + n;

    const float* Tb  = Tbuf + (size_t)b * NNODES * 16;
    const float  pd0 = pden[0], pd1 = pden[1], pd2 = pden[2], pd3 = pden[3];

    // rolling pll, pre-selected into B-operand layout (this half's K pair)
    float rAx = 0.f, rAy = 0.f, rBx = 0.f, rBy = 0.f;
    float accA = 0.f, accB = 0.f;                 // log2 accumulators
    float uA0 = 1.f, uA1 = 1.f, uA2 = 1.f, uA3 = 1.f;
    float uB0 = 1.f, uB1 = 1.f, uB2 = 1.f, uB3 = 1.f;
    float ivA = 1.f, ivB = 1.f;

    // ---- topology pipeline: cur = step i, nxt = step i+1 ------------------
    int cm0 = RFL(child_mask[0]), cm1 = RFL(child_mask[1]), cm2 = RFL(child_mask[2]);
    int cc0 = RFL(children[b * 3 + 0]);
    int cc1 = RFL(children[b * 3 + 1]);
    int cc2 = RFL(children[b * 3 + 2]);
    int cpar = RFL(parent[b]);
    const int j1 = (NINT > 1) ? 1 : 0;
    int nm0 = RFL(child_mask[j1 * 3 + 0]), nm1 = RFL(child_mask[j1 * 3 + 1]),
        nm2 = RFL(child_mask[j1 * 3 + 2]);
    int nc0 = RFL(children[(j1 * BS + b) * 3 + 0]);
    int nc1 = RFL(children[(j1 * BS + b) * 3 + 1]);
    int nc2 = RFL(children[(j1 * BS + b) * 3 + 2]);
    int npar = RFL(parent[j1 * BS + b]);

    // ---- vector prefetch state (A operand + tip B operands per child) -----
    v2f am[3], tba[3], tbb[3];
    int rol[3] = { 0, 0, 0 };
    auto pref = [&](int msk, int idx, int rollid, int c) {
        if (!msk) return;                          // scalar branch
        const float* tp = Tb + idx * 16;           // uniform base (saddr)
        v2f a = { tp[offT], tp[offT + 1] };        // b64 @ voffset
        am[c] = a;
        const int r = (idx == rollid);
        rol[c] = r;
        if (!r) {                                  // tip: ll[k][site], k=kb,kb+1
            const float* lp = L + (size_t)idx * (4 * SEQLEN); // uniform base
            v2f xa = { lp[offLA], lp[offLA + SEQLEN] };
            v2f xb = { lp[offLB], lp[offLB + SEQLEN] };
            tba[c] = xa;  tbb[c] = xb;
        }
    };
    pref(cm0, cc0, -1, 0);                         // step 0 vectors in flight
    pref(cm1, cc1, -1, 1);
    pref(cm2, cc2, -1, 2);

    for (int i = 0; i < NINT; ++i) {
        // ---- compute step i from prefetched operands ----------------------
        uA0 = 1.f; uA1 = 1.f; uA2 = 1.f; uA3 = 1.f;
        uB0 = 1.f; uB1 = 1.f; uB2 = 1.f; uB3 = 1.f;
#pragma unroll
        for (int c = 0; c < 3; ++c) {
            const int mk = (c == 0) ? cm0 : ((c == 1) ? cm1 : cm2);
            if (!mk) continue;                     // scalar branch: EXEC all-1
            v2f amat = am[c];
            v2f bmA, bmB;
            if (rol[c]) {                          // rolling pll, already in layout
                bmA.x = rAx; bmA.y = rAy;
                bmB.x = rBx; bmB.y = rBy;
            } else {
                bmA = tba[c];  bmB = tbb[c];
            }
            v8f z = {};
            v8f dA = __builtin_amdgcn_wmma_f32_16x16x4_f32(
                false, amat, false, bmA, (short)0, z, false, false);
            v8f dB = __builtin_amdgcn_wmma_f32_16x16x4_f32(
                false, amat, false, bmB, (short)0, z, false, false);
            if (c == 0) {                          // assign, don't multiply
                uA0 = dA[0]; uA1 = dA[1]; uA2 = dA[2]; uA3 = dA[3];
                uB0 = dB[0]; uB1 = dB[1]; uB2 = dB[2]; uB3 = dB[3];
            } else {
                uA0 *= dA[0]; uA1 *= dA[1]; uA2 *= dA[2]; uA3 *= dA[3];
                uB0 *= dB[0]; uB1 *= dB[1]; uB2 *= dB[2]; uB3 *= dB[3];
            }
        }

        // ---- advance pipelines (loads overlap the VALU tail below) --------
        const int rollid = cpar;                   // parent(i) = rolling child of i+1
        cm0 = nm0; cm1 = nm1; cm2 = nm2;
        cc0 = nc0; cc1 = nc1; cc2 = nc2; cpar = npar;
        pref(cm0, cc0, rollid, 0);                 // vectors for step i+1
        pref(cm1, cc1, rollid, 1);
        pref(cm2, cc2, rollid, 2);
        const int j2 = (i + 2 < NINT) ? i + 2 : NINT - 1;
        nm0 = RFL(child_mask[j2 * 3 + 0]);
        nm1 = RFL(child_mask[j2 * 3 + 1]);
        nm2 = RFL(child_mask[j2 * 3 + 2]);
        nc0 = RFL(children[(j2 * BS + b) * 3 + 0]);
        nc1 = RFL(children[(j2 * BS + b) * 3 + 1]);
        nc2 = RFL(children[(j2 * BS + b) * 3 + 2]);
        npar = RFL(parent[j2 * BS + b]);

        // ---- lane-local scaler / log2; rolling operand built directly -----
        float scA = (uA0 + uA1) + (uA2 + uA3);
        ivA = __builtin_amdgcn_rcpf(scA);          // v_rcp_f32
        accA += __builtin_amdgcn_logf(scA);        // v_log_f32 (log2)
        rAx = (h ? uA2 : uA0) * ivA;               // B layout: this half's K pair
        rAy = (h ? uA3 : uA1) * ivA;

        float scB = (uB0 + uB1) + (uB2 + uB3);
        ivB = __builtin_amdgcn_rcpf(scB);
        accB += __builtin_amdgcn_logf(scB);
        rBx = (h ? uB2 : uB0) * ivB;
        rBy = (h ? uB3 : uB1) * ivB;
    }

    // root term (pden . root_pll from final u,iv), log2 -> ln, site weighting
    float rsA = ((pd0 * uA0 + pd1 * uA1) + (pd2 * uA2 + pd3 * uA3)) * ivA;
    float rsB = ((pd0 * uB0 + pd1 * uB1) + (pd2 * uB2 + pd3 * uB3)) * ivB;
    accA += __builtin_amdgcn_logf(rsA);
    accB += __builtin_amdgcn_logf(rsB);
    accA *= LN2F * site_counts[sA + n];
    accB *= LN2F * site_counts[sB + n];

    // butterfly sum over the 16 sites (masks 1..8 stay within lane halves)
#pragma unroll
    for (int mds = 8; mds >= 1; mds >>= 1) {
        accA += __shfl_xor(accA, mds, 32);
        accB += __shfl_xor(accB, mds, 32);
    }
    if (lane == 0) {
        partials[b * TSLOTS + tA] = accA;
        if (hasB) partials[b * TSLOTS + tB] = accB;
    }
}

// ---------------------------------------------------------------------------
// Kernel 3: deterministic tree reduction of 125 tile partials per batch.
// ---------------------------------------------------------------------------
__global__ void phy_reduce(const float* __restrict__ partials,
                           float* __restrict__ out) {
    const int b = blockIdx.x;
    const int t = threadIdx.x;               // 128 threads
    __shared__ float sm[128];
    sm[t] = (t < TILES) ? partials[b * TSLOTS + t] : 0.f;
    __syncthreads();
    for (int off = 64; off > 0; off >>= 1) {
        if (t < off) sm[t] += sm[t + off];
        __syncthreads();
    }
    if (t == 0) out[b] = sm[0];
}

// ---------------------------------------------------------------------------
extern "C" void kernel_launch(void* const* d_in, const int* in_sizes, int n_in,
                              void* d_out, int out_size, void* d_ws, size_t ws_size,
                              hipStream_t stream) {
    (void)in_sizes; (void)n_in; (void)out_size; (void)ws_size;
    const float* log_branch  = (const float*)d_in[0];
    const float* L           = (const float*)d_in[1];
    const float* site_counts = (const float*)d_in[2];
    const float* pden        = (const float*)d_in[3];
    const float* D4          = (const float*)d_in[4];
    const float* U           = (const float*)d_in[5];
    const float* Uinv        = (const float*)d_in[6];
    const int*   children    = (const int*)d_in[7];
    const unsigned char* child_mask = (const unsigned char*)d_in[8];
    const int*   parent      = (const int*)d_in[9];

    float* Tbuf     = (float*)d_ws;                                       // 520192 B
    float* partials = (float*)((char*)d_ws +
                               (size_t)BS * NNODES * 16 * sizeof(float)); // 16 KB

    phy_Tprep<<<dim3((BS * NNODES + 255) / 256), dim3(256), 0, stream>>>(
        log_branch, D4, U, Uinv, Tbuf);
    phy_scan<<<dim3((NPAIRS + 7) / 8, BS), dim3(256), 0, stream>>>(
        Tbuf, L, site_counts, pden, children, child_mask, parent, partials);
    phy_reduce<<<dim3(BS), dim3(128), 0, stream>>>(partials, (float*)d_out);
}